// GeneralGaussianBlur3D_notperiodic_1271310320307
// MI455X (gfx1250) — compile-verified
//
#include <hip/hip_runtime.h>
#include <hip/hip_bf16.h>

typedef __attribute__((ext_vector_type(2))) float v2f;
typedef __attribute__((ext_vector_type(8))) float v8f;

#define NV 256
#define WIN 176          // A window columns per 128-wide j-block (128 + 2*24)
#define APITCH 180       // WIN + 2 corner cols + pad; 180%64=52, gcd(52,64)=4 -> conflict-free
#define KSUB 64          // per-16-col-tile K window (margin 24: exp(-24^2/18) ~ 1e-14)

// ---------------------------------------------------------------------------
// Exact per-axis blur matrices B[k][y] (256x256, row-major) + wrap-mass vectors.
// axis 0: P=274, pad=9, sigma=3 (X and Y passes);  axis 1: P=268, pad=6, sigma=2 (Z).
// Column sums are exactly 1; circular-wrap mass is also recorded separately in
// wUp[y] (wrapped to k~255, nonzero only for small y) and wDn[y] (to k~0, large y).
// ---------------------------------------------------------------------------
__global__ __launch_bounds__(256) void build_B_kernel(float* B0, float* B1, float* wv) {
  const int axis = blockIdx.x;
  float* Bm        = (axis == 0) ? B0 : B1;
  float* wUp       = wv + axis * 512;
  float* wDn       = wUp + 256;
  const int   P    = (axis == 0) ? 274 : 268;
  const int   pad  = (axis == 0) ? 9 : 6;
  const float i2s2 = (axis == 0) ? (1.0f / 18.0f) : (1.0f / 8.0f);
  const int   N    = P / 2;
  const int   y    = threadIdx.x;

  for (int j = 0; j < NV; ++j) Bm[j * NV + y] = 0.0f;
  wUp[y] = 0.0f; wDn[y] = 0.0f;

  float S = 0.0f;
  for (int ti = 0; ti < P; ++ti) {
    float t = (float)(ti - N);
    S += expf(-t * t * i2s2);
  }
  const float invS = 1.0f / S;

  for (int ti = 0; ti < P; ++ti) {
    int   t = ti - N;
    float w = expf(-(float)(t * t) * i2s2) * invS;
    int praw = y + pad - t;                 // |praw| < 2P
    int p = praw; p %= P; if (p < 0) p += P;
    int j = p - pad;
    j = j < 0 ? 0 : (j > NV - 1 ? NV - 1 : j);
    Bm[j * NV + y] += w;
    if      (praw < 0)  wUp[y] += w;        // circular wrap to the high end
    else if (praw >= P) wDn[y] += w;        // circular wrap to the low end
  }
}

// ---------------------------------------------------------------------------
// Banded GEMM pass: out[q, j] = sum_{k in window} A[q,k]*B[k,j]  + wrap corrections
//   volume address of (q, j) = base(q) + j*STRIDE
//   PASS 0 (Y): base=q*256, stride=1 | PASS 1 (X): base=(q>>8)<<16|(q&255), stride=256
//   PASS 2 (Z): base=q, stride=65536
// Block: 256 threads / 8 waves; tile 16 q x 128 j; per 16-col tile: 16 x wmma_f32_16x16x4.
// ---------------------------------------------------------------------------
template <int PASS>
__global__ __launch_bounds__(256) void blur_pass_kernel(const float* __restrict__ in,
                                                        const float* __restrict__ Bm,
                                                        const float* __restrict__ wUp,
                                                        const float* __restrict__ wDn,
                                                        float* __restrict__ out) {
  __shared__ float At[16][APITCH];          // cols [0,WIN): window; 176:k=0, 177:k=255
  __shared__ float Ct[128][17];

  const int tid = threadIdx.x;
  const int q0  = (blockIdx.x >> 1) << 4;
  const int jg  = blockIdx.x & 1;
  const int j0  = jg << 7;
  const int w0  = jg ? 80 : 0;              // window start: [0,176) or [80,256)

  // ---- stage A window (16 x 176) + corner columns into LDS ----
  if (PASS == 0) {
    for (int i = 0; i < 11; ++i) {
      int s  = tid + (i << 8);              // 0..2815
      int qL = s / WIN;
      int k  = s - qL * WIN;
      At[qL][k] = in[(size_t)(q0 + qL) * NV + (w0 + k)];
    }
    if (tid < 32) {
      int qL = tid & 15, which = tid >> 4;  // 0 -> k=0, 1 -> k=255
      At[qL][WIN + which] = in[(size_t)(q0 + qL) * NV + (which ? 255 : 0)];
    }
  } else {
    const size_t str = (PASS == 1) ? 256u : 65536u;
    for (int i = 0; i < 11; ++i) {
      int s  = tid + (i << 8);
      int qL = s & 15;
      int kk = s >> 4;                      // 0..175
      int q  = q0 + qL;
      size_t base = (PASS == 1) ? (((size_t)(q >> 8) << 16) + (size_t)(q & 255))
                                : (size_t)q;
      At[qL][kk] = in[base + (size_t)(w0 + kk) * str];
    }
    if (tid < 32) {
      int qL = tid & 15, which = tid >> 4;
      int q  = q0 + qL;
      size_t base = (PASS == 1) ? (((size_t)(q >> 8) << 16) + (size_t)(q & 255))
                                : (size_t)q;
      At[qL][WIN + which] = in[base + (size_t)(which ? 255 : 0) * str];
    }
  }
  __syncthreads();

  const int wave = tid >> 5;
  const int lane = tid & 31;
  const int m    = lane & 15;
  const int hk   = (lane >> 4) << 1;        // 0 / 2
  const int mh   = (lane >> 4) << 3;        // 0 / 8
  const int jw   = j0 + (wave << 4);

  // per-tile K sub-window [s0, s0+63], even-aligned, inside the block window
  int s0 = jw - 24;
  {
    const int lo = w0, hi = w0 + WIN - KSUB;
    s0 = s0 < lo ? lo : (s0 > hi ? hi : s0);
  }

  v8f c = {};
  for (int it = 0; it < KSUB / 4; ++it) {
    const int k0 = s0 + (it << 2);
    v2f a = *(const v2f*)&At[m][(k0 - w0) + hk];            // A 16x4 frag (8B aligned)
    const float* bp = Bm + (size_t)(k0 + hk) * NV + (jw + m); // B 4x16 frag (L2 resident)
    v2f b; b.x = bp[0]; b.y = bp[NV];
    c = __builtin_amdgcn_wmma_f32_16x16x4_f32(
        false, a, false, b, (short)0, c, false, false);
  }

  // ---- exact circular-wrap corner corrections ----
  {
    const float cu = wUp[jw + m];           // mass that wrapped to k~255
    const float cd = wDn[jw + m];           // mass that wrapped to k~0
    for (int r = 0; r < 8; ++r) {
      int M = r + mh;
      c[r] += cu * At[M][WIN + 1] + cd * At[M][WIN + 0];
    }
  }

  // ---- store, coalesced along each pass's fast axis ----
  if (PASS == 0) {
    for (int r = 0; r < 8; ++r) {
      int q = q0 + r + mh;
      out[(size_t)q * NV + (jw + m)] = c[r];
    }
  } else {
    for (int r = 0; r < 8; ++r) {
      Ct[(wave << 4) + m][r + mh] = c[r];
    }
    __syncthreads();
    for (int i = 0; i < 8; ++i) {
      int s  = tid + (i << 8);
      int qL = s & 15;
      int jL = s >> 4;
      int q  = q0 + qL;
      int j  = j0 + jL;
      size_t addr = (PASS == 1)
          ? (((size_t)(q >> 8) << 16) + (size_t)j * NV + (size_t)(q & 255))
          : ((size_t)j * 65536u + (size_t)q);
      out[addr] = Ct[jL][qL];
    }
  }
}

// ---------------------------------------------------------------------------
extern "C" void kernel_launch(void* const* d_in, const int* in_sizes, int n_in,
                              void* d_out, int out_size, void* d_ws, size_t ws_size,
                              hipStream_t stream) {
  const float* u  = (const float*)d_in[0];
  float* outp     = (float*)d_out;
  char*  ws       = (char*)d_ws;

  float* B0 = (float*)ws;                           // 256 KB (sigma=3, X/Y)
  float* B1 = (float*)(ws + (size_t)(256 << 10));   // 256 KB (sigma=2, Z)
  float* wv = (float*)(ws + (size_t)(512 << 10));   // 4 x 256 wrap-mass vectors
  float* T  = (float*)(ws + (size_t)(1 << 20));     // 64 MB intermediate

  build_B_kernel<<<dim3(2), dim3(256), 0, stream>>>(B0, B1, wv);

  const dim3 grid(8192), block(256);
  float* wUpA = wv, *wDnA = wv + 256, *wUpZ = wv + 512, *wDnZ = wv + 768;

  // reference order: blur Y (axis 2), then X (axis 1), then Z (axis 0)
  blur_pass_kernel<0><<<grid, block, 0, stream>>>(u,    B0, wUpA, wDnA, outp); // Y
  blur_pass_kernel<1><<<grid, block, 0, stream>>>(outp, B0, wUpA, wDnA, T);    // X
  blur_pass_kernel<2><<<grid, block, 0, stream>>>(T,    B1, wUpZ, wDnZ, outp); // Z
}